// Net_36593121362456
// MI455X (gfx1250) — compile-verified
//
#include <hip/hip_runtime.h>
#include <hip/hip_bf16.h>

typedef __attribute__((ext_vector_type(2))) float v2f;
typedef __attribute__((ext_vector_type(8))) float v8f;

#define HID 16
#define NUM_GRAPHS 64

// ---------------------------------------------------------------------------
// Degree / normalization
// ---------------------------------------------------------------------------
__global__ __launch_bounds__(256) void count_deg_kernel(const int* __restrict__ dst,
                                                        float* __restrict__ deg,
                                                        int n_edges) {
    int e = blockIdx.x * blockDim.x + threadIdx.x;
    if (e < n_edges) atomicAdd(&deg[dst[e]], 1.0f);
}

__global__ __launch_bounds__(256) void deg_to_dis_kernel(float* __restrict__ deg, int n) {
    int i = blockIdx.x * blockDim.x + threadIdx.x;
    if (i < n) deg[i] = rsqrtf(deg[i] + 1.0f);   // self-loop contributes +1
}

// ---------------------------------------------------------------------------
// WMMA fp32 GEMM:  out[n,16] = A[n,K] @ W[K,16]
// 128 threads = 4 waves; each wave computes one 16x16 tile via 16x16x4 f32 WMMA.
// A tile (64 x K) and W (K x 16) staged through LDS with coalesced float4 loads.
//
// Fragment layout (CDNA5 ISA 7.12.2):
//   A (16x4 f32, 2 VGPR): lane L holds row M=L%16; lanes 0-15 -> K={0,1},
//                         lanes 16-31 -> K={2,3}  (VGPR0=K even, VGPR1=K odd)
//   B (4x16 f32, 2 VGPR): lane L holds col N=L%16; lanes 0-15 -> K={0,1},
//                         lanes 16-31 -> K={2,3}
//   C/D (16x16 f32, 8 VGPR): acc[v] = element (M = v + (L>=16 ? 8 : 0), N = L%16)
// ---------------------------------------------------------------------------
template <int K>
__global__ __launch_bounds__(128) void gemm_wmma_kernel(const float* __restrict__ A,
                                                        const float* __restrict__ W,
                                                        float* __restrict__ out,
                                                        int n) {
    constexpr int ROWS = 64;               // 4 waves * 16 rows
    __shared__ float sA[ROWS * K];
    __shared__ float sB[K * HID];

    const int tid  = threadIdx.x;
    const int row0 = blockIdx.x * ROWS;

    // Stage W (K x 16) into LDS
    for (int i = tid; i < K * HID; i += 128) sB[i] = W[i];

    // Stage A tile (ROWS x K) into LDS, coalesced float4, zero-pad OOB rows
    for (int i = tid * 4; i < ROWS * K; i += 128 * 4) {
        int r = i / K;
        int c = i - r * K;
        int row = row0 + r;
        float4 v = {0.f, 0.f, 0.f, 0.f};
        if (row < n) v = *(const float4*)(A + (size_t)row * K + c);
        *(float4*)(sA + i) = v;
    }
    __syncthreads();

    const int lane = tid & 31;
    const int wave = tid >> 5;
    const int mrow = lane & 15;            // M (for A) / N (for B)
    const int koff = (lane >> 4) << 1;     // 0 for lanes 0-15, 2 for lanes 16-31
    const float* aRow = sA + (wave * 16 + mrow) * K;

    v8f acc = {};
#pragma unroll
    for (int k0 = 0; k0 < K; k0 += 4) {
        v2f a, b;
        a.x = aRow[k0 + koff];
        a.y = aRow[k0 + koff + 1];
        b.x = sB[(k0 + koff) * HID + mrow];
        b.y = sB[(k0 + koff + 1) * HID + mrow];
        acc = __builtin_amdgcn_wmma_f32_16x16x4_f32(
            /*neg_a=*/false, a, /*neg_b=*/false, b,
            /*c_mod=*/(short)0, acc, /*reuse_a=*/false, /*reuse_b=*/false);
    }

    const int col   = mrow;
    const int rbase = row0 + wave * 16 + ((lane >> 4) << 3);
#pragma unroll
    for (int v = 0; v < 8; ++v) {
        int row = rbase + v;
        if (row < n) out[(size_t)row * HID + col] = acc[v];
    }
}

// ---------------------------------------------------------------------------
// Edge scatter: agg[dst] += h[src] * dis[src] * dis[dst]
// One thread per (edge, feature); h/agg/dis working set fits in L2 (192 MB),
// so the f32 atomics resolve at L2 rate.
// ---------------------------------------------------------------------------
__global__ __launch_bounds__(256) void scatter_kernel(const int* __restrict__ src,
                                                      const int* __restrict__ dst,
                                                      const float* __restrict__ dis,
                                                      const float* __restrict__ h,
                                                      float* __restrict__ agg,
                                                      int n_edges) {
    int t = blockIdx.x * blockDim.x + threadIdx.x;
    if (t >= n_edges * HID) return;
    int e = t >> 4;
    int f = t & 15;
    int s = src[e];
    int d = dst[e];
    float norm = dis[s] * dis[d];
    atomicAdd(&agg[d * HID + f], h[s * HID + f] * norm);
}

// ---------------------------------------------------------------------------
// Finalize layer 1 (in place into agg): h = relu(agg + dis^2 * hpre + b)
// ---------------------------------------------------------------------------
__global__ __launch_bounds__(256) void finalize_kernel(float* __restrict__ agg,
                                                       const float* __restrict__ hpre,
                                                       const float* __restrict__ dis,
                                                       const float* __restrict__ bias,
                                                       int n) {
    int t = blockIdx.x * blockDim.x + threadIdx.x;
    if (t >= n * HID) return;
    int i = t >> 4;
    int f = t & 15;
    float di = dis[i];
    float v = agg[t] + di * di * hpre[t] + bias[f];
    agg[t] = fmaxf(v, 0.f);
}

// ---------------------------------------------------------------------------
// Finalize layer 2 fused with global_add_pool into pooled[64,16]
// ---------------------------------------------------------------------------
__global__ __launch_bounds__(256) void finalize_pool_kernel(const float* __restrict__ agg,
                                                            const float* __restrict__ hpre,
                                                            const float* __restrict__ dis,
                                                            const float* __restrict__ bias,
                                                            const int* __restrict__ batch,
                                                            float* __restrict__ pooled,
                                                            int n) {
    int t = blockIdx.x * blockDim.x + threadIdx.x;
    if (t >= n * HID) return;
    int i = t >> 4;
    int f = t & 15;
    float di = dis[i];
    float v = fmaxf(agg[t] + di * di * hpre[t] + bias[f], 0.f);
    atomicAdd(&pooled[batch[i] * HID + f], v);
}

// ---------------------------------------------------------------------------
// Head: out[g] = sigmoid(pooled[g,:] . fc_w + fc_b)
// ---------------------------------------------------------------------------
__global__ __launch_bounds__(64) void head_kernel(const float* __restrict__ pooled,
                                                  const float* __restrict__ fc_w,
                                                  const float* __restrict__ fc_b,
                                                  float* __restrict__ out) {
    int g = threadIdx.x;
    if (g < NUM_GRAPHS) {
        float s = fc_b[0];
#pragma unroll
        for (int f = 0; f < HID; ++f) s += pooled[g * HID + f] * fc_w[f];
        out[g] = 1.0f / (1.0f + expf(-s));
    }
}

// ---------------------------------------------------------------------------
extern "C" void kernel_launch(void* const* d_in, const int* in_sizes, int n_in,
                              void* d_out, int out_size, void* d_ws, size_t ws_size,
                              hipStream_t stream) {
    const float* x     = (const float*)d_in[0];
    const int*   edge  = (const int*)d_in[1];
    const int*   batch = (const int*)d_in[2];
    const float* W1    = (const float*)d_in[3];
    const float* b1    = (const float*)d_in[4];
    const float* W2    = (const float*)d_in[5];
    const float* b2    = (const float*)d_in[6];
    const float* fcw   = (const float*)d_in[7];
    const float* fcb   = (const float*)d_in[8];
    float*       out   = (float*)d_out;

    const int F_IN    = 128;
    const int n_nodes = in_sizes[0] / F_IN;
    const int n_edges = in_sizes[1] / 2;
    const int* srcIdx = edge;              // edge_index[0]
    const int* dstIdx = edge + n_edges;    // edge_index[1]

    // Workspace layout (floats): dis | bufA (h1pre->h2pre) | bufB (agg1->h1->agg2) | pooled
    float* ws = (float*)d_ws;
    size_t off = 0;
    float* dis = ws + off;   off += ((size_t)n_nodes + 127) & ~(size_t)127;
    float* bufA = ws + off;  off += (size_t)n_nodes * HID;
    float* bufB = ws + off;  off += (size_t)n_nodes * HID;
    float* pooled = ws + off;

    const int B256 = 256;
    const int gEdges  = (n_edges + B256 - 1) / B256;
    const int gNodes  = (n_nodes + B256 - 1) / B256;
    const int gNF     = (n_nodes * HID + B256 - 1) / B256;
    const int gEF     = (n_edges * HID + B256 - 1) / B256;
    const int gTiles  = (n_nodes + 63) / 64;

    // 1) degrees -> dis = rsqrt(deg+1)
    hipMemsetAsync(dis, 0, (size_t)n_nodes * sizeof(float), stream);
    count_deg_kernel<<<gEdges, B256, 0, stream>>>(dstIdx, dis, n_edges);
    deg_to_dis_kernel<<<gNodes, B256, 0, stream>>>(dis, n_nodes);

    // 2) layer 1: h1pre = x @ W1   (WMMA f32 16x16x4, K=128)
    gemm_wmma_kernel<128><<<gTiles, 128, 0, stream>>>(x, W1, bufA, n_nodes);

    // 3) agg1 = scatter; h1 = relu(agg1 + dis^2*h1pre + b1) in place in bufB
    hipMemsetAsync(bufB, 0, (size_t)n_nodes * HID * sizeof(float), stream);
    scatter_kernel<<<gEF, B256, 0, stream>>>(srcIdx, dstIdx, dis, bufA, bufB, n_edges);
    finalize_kernel<<<gNF, B256, 0, stream>>>(bufB, bufA, dis, b1, n_nodes);

    // 4) layer 2: h2pre = h1 @ W2  (WMMA f32 16x16x4, K=16) -> overwrite bufA
    gemm_wmma_kernel<16><<<gTiles, 128, 0, stream>>>(bufB, W2, bufA, n_nodes);

    // 5) agg2 = scatter (bufB reused); finalize + pool
    hipMemsetAsync(bufB, 0, (size_t)n_nodes * HID * sizeof(float), stream);
    scatter_kernel<<<gEF, B256, 0, stream>>>(srcIdx, dstIdx, dis, bufA, bufB, n_edges);
    hipMemsetAsync(pooled, 0, (size_t)NUM_GRAPHS * HID * sizeof(float), stream);
    finalize_pool_kernel<<<gNF, B256, 0, stream>>>(bufB, bufA, dis, b2, batch, pooled, n_nodes);

    // 6) sigmoid head -> d_out [64]
    head_kernel<<<1, 64, 0, stream>>>(pooled, fcw, fcb, out);
}